// MCPBRNN_SW_Variant_Routing_10196252360856
// MI455X (gfx1250) — compile-verified
//
#include <hip/hip_runtime.h>
#include <hip/hip_bf16.h>

// ---------------------------------------------------------------------------
// MCPBRNN (H=1): one flat scalar nonlinear recurrence of length (B-tl)*S.
// Serial-latency bound; single wave runs the chain, CDNA5 TDM (tensor_load_
// to_lds) double-buffers x into LDS, s_wait_tensorcnt synchronizes.
// Gate sigmoid computed via V_TANH_F32 (one trans op) when available.
// ---------------------------------------------------------------------------

#define CHUNK 2048  // floats per TDM tile: 8 KB, double buffered -> 16 KB LDS

typedef unsigned int u32x4 __attribute__((ext_vector_type(4)));
typedef int          i32x8 __attribute__((ext_vector_type(8)));
typedef int          i32x4 __attribute__((ext_vector_type(4)));

// 1-D TDM load: `elems` fp32 from global `gptr` -> LDS byte offset `lds_off`.
// D# group0: count=1 | lds_addr | global_addr[56:0] | type=2.
// D# group1: data_size=4B, tensor_dim0=elems, tensor_dim1=1, tile_dim0=elems,
//            tile_dim1/2=0 (1-D), dim0_stride=elems, no multicast/pad/iterate.
__device__ __forceinline__ void tdm_load_1d(unsigned lds_off, const void* gptr,
                                            int elems) {
#if defined(__HIP_DEVICE_COMPILE__)
  unsigned long long ga = (unsigned long long)gptr;
  u32x4 g0 = {1u,                                  // count=1 (valid), user desc
              lds_off,                             // lds_addr (bytes)
              (unsigned)ga,                        // global_addr[31:0]
              ((unsigned)(ga >> 32) & 0x01FFFFFFu) | (2u << 30)};  // [56:32]|type=2
  unsigned e = (unsigned)elems;
  i32x8 g1;
  g1[0] = (int)(2u << 16);                  // wg_mask=0, data_size=2 (4 bytes)
  g1[1] = (int)((e & 0xFFFFu) << 16);       // tensor_dim0[15:0] @ bits 63:48
  g1[2] = (int)((e >> 16) | (1u << 16));    // tensor_dim0[31:16], tensor_dim1=1
  g1[3] = (int)(e << 16);                   // tensor_dim1 hi=0, tile_dim0=elems
  g1[4] = 0;                                // tile_dim1=0, tile_dim2=0 (1-D)
  g1[5] = (int)e;                           // tensor_dim0_stride[31:0]
  g1[6] = 0;                                // stride hi, dim1_stride lo
  g1[7] = 0;
  i32x4 z4 = {0, 0, 0, 0};                  // groups 2/3 unused (<=2-D tensor)
  i32x8 z8 = {0, 0, 0, 0, 0, 0, 0, 0};      // extra group (clang-23 6-arg form)
  __builtin_amdgcn_tensor_load_to_lds(g0, g1, z4, z4, z8, 0);
#else
  (void)lds_off; (void)gptr; (void)elems;
#endif
}

__device__ __forceinline__ void wait_tensorcnt_le(int n) {
#if defined(__HIP_DEVICE_COMPILE__)
  if (n == 0) __builtin_amdgcn_s_wait_tensorcnt(0);
  else        __builtin_amdgcn_s_wait_tensorcnt(1);
#else
  (void)n;
#endif
}

__device__ __forceinline__ float fast_rcp(float v) {
#if defined(__HIP_DEVICE_COMPILE__)
  return __builtin_amdgcn_rcpf(v);
#else
  return 1.0f / v;
#endif
}

// oo = A * tanh(fma(c, ah, bz)) + A  when V_TANH_F32 is available (1 trans op),
// else oo = oo1 / (1 + exp(-oo2))                          (2 trans ops).
#if defined(__HIP_DEVICE_COMPILE__) && __has_builtin(__builtin_amdgcn_tanhf)
#define GATE_USES_TANH 1
__device__ __forceinline__ float gate_oo(float c, float ah, float bz, float A) {
  return __builtin_fmaf(__builtin_amdgcn_tanhf(__builtin_fmaf(c, ah, bz)), A, A);
}
#else
#define GATE_USES_TANH 0
__device__ __forceinline__ float gate_oo(float c, float ah, float bz, float A) {
  // here ah/bz/A are set up so that oo2 = 2*(c*ah + bz), oo1 = 2*A
  const float oo2 = 2.0f * __builtin_fmaf(c, ah, bz);
  return (2.0f * A) * fast_rcp(1.0f + __expf(-oo2));
}
#endif

__global__ __launch_bounds__(32) void mcpbrnn_routing_kernel(
    const float* __restrict__ x, const int* __restrict__ tlp,
    const float* __restrict__ p_mean, const float* __restrict__ p_std,
    const float* __restrict__ w_r_yom, const float* __restrict__ w_r_yfm,
    const float* __restrict__ b0p, const float* __restrict__ w_b1,
    float* __restrict__ out, int B, int S) {
  __shared__ float buf[2][CHUNK];
  const int tid = (int)threadIdx.x;

  // Rows < time_lag stay zero; everything is overwritten for rows >= tl.
  for (int j = tid; j < 4 * B; j += 32) out[j] = 0.0f;

  const int tl = tlp[0];
  // Fold parameters.  oo2 = (c - mo)*a + b0 with a = w1/so.
  // sigmoid(oo2) = 0.5*(1 + tanh(oo2/2)):
  //   z  = c*ah + bz,   ah = a/2,  bz = b0/2 - mo*a/2
  //   oo = A*tanh(z) + A,          A  = oo1/2
  const float mo  = p_mean[0];
  const float so  = p_std[0];
  const float eo  = __expf(w_r_yom[0]);
  const float ef  = __expf(w_r_yfm[0]);
  const float oo1 = eo * fast_rcp(eo + ef);
  const float b0v = b0p[0];
  const float a   = w_b1[0] * fast_rcp(so);
  const float ah  = 0.5f * a;
  const float bz  = 0.5f * b0v - mo * ah;
  const float A   = 0.5f * oo1;

  const float* xs = x + (long long)tl * S;
  const long long N = (long long)(B - tl) * S;          // flat recurrence length
  const int nchunks = (int)((N + CHUNK - 1) / CHUNK);

  const unsigned lds0 = (unsigned)(unsigned long long)(&buf[0][0]);
  const unsigned lds1 = (unsigned)(unsigned long long)(&buf[1][0]);

  auto chunk_cnt = [&](int k) -> int {
    long long rem = N - (long long)k * CHUNK;
    return (int)(rem < (long long)CHUNK ? rem : (long long)CHUNK);
  };

  // Prologue: two TDM tiles in flight (TENSORcnt = 2).
  if (nchunks > 0) tdm_load_1d(lds0, xs, chunk_cnt(0));
  if (nchunks > 1) tdm_load_1d(lds1, xs + CHUNK, chunk_cnt(1));

  float c = 0.0f;   // carry (h never feeds back; only c matters)
  int   t = 0;      // position within row
  int   b = tl;     // current row

  for (int k = 0; k < nchunks; ++k) {
    // TDM completes in-order per wave: cnt<=1 -> oldest (this chunk) is done.
    if (k + 1 < nchunks) wait_tensorcnt_le(1);
    else                 wait_tensorcnt_le(0);

    const int cnt = chunk_cnt(k);
    if (tid == 0) {
      const float* __restrict__ u = &buf[k & 1][0];
#pragma unroll 4
      for (int j = 0; j < cnt; ++j) {
        const float uj = u[j];
        const float oo = gate_oo(c, ah, bz, A);          // fma -> tanh -> fma
        if (t == S - 1) {                                // record pre-update state
          out[b]         = oo * c;                       // h0_out
          out[B + b]     = c;                            // c0
          out[2 * B + b] = oo;                           // oo
          out[3 * B + b] = 1.0f - oo;                    // f
        }
        c = __builtin_fmaf(-oo, c, c) + uj;              // c1 = (1-oo)*c0 + u
        if (++t == S) { t = 0; ++b; }
      }
    }
    // Refill the buffer just drained with chunk k+2 (same parity).
    if (k + 2 < nchunks)
      tdm_load_1d((k & 1) ? lds1 : lds0,
                  xs + (long long)(k + 2) * CHUNK, chunk_cnt(k + 2));
  }
}

extern "C" void kernel_launch(void* const* d_in, const int* in_sizes, int n_in,
                              void* d_out, int out_size, void* d_ws, size_t ws_size,
                              hipStream_t stream) {
  (void)n_in; (void)d_ws; (void)ws_size;
  // setup_inputs order:
  // 0:x 1:epoch 2:time_lag 3:y_obs 4:p_mean 5:p_std
  // 6:weight_r_yom 7:weight_r_yfm 8:bias_b0_yom 9:weight_b1_yom
  const float* x       = (const float*)d_in[0];
  const int*   tl      = (const int*)d_in[2];
  const float* p_mean  = (const float*)d_in[4];
  const float* p_std   = (const float*)d_in[5];
  const float* w_r_yom = (const float*)d_in[6];
  const float* w_r_yfm = (const float*)d_in[7];
  const float* b0      = (const float*)d_in[8];
  const float* w_b1    = (const float*)d_in[9];
  float* out = (float*)d_out;

  const int H = in_sizes[8];              // == 1 for this model
  const int B = out_size / (4 * H);       // 4 outputs of [B,H] concatenated
  const int S = in_sizes[0] / B;          // x is [B,S]

  mcpbrnn_routing_kernel<<<dim3(1), dim3(32), 0, stream>>>(
      x, tl, p_mean, p_std, w_r_yom, w_r_yfm, b0, w_b1, out, B, S);
}